// STNDT_9397388443695
// MI455X (gfx1250) — compile-verified
//
#include <hip/hip_runtime.h>

typedef __attribute__((ext_vector_type(2))) float v2f;
typedef __attribute__((ext_vector_type(8))) float v8f;

__device__ __forceinline__ v8f wmma_f32_k4(v2f a, v2f b, v8f c) {
  // D = A(16x4 f32) * B(4x16 f32) + C(16x16 f32)
  return __builtin_amdgcn_wmma_f32_16x16x4_f32(
      /*neg_a=*/false, a, /*neg_b=*/false, b,
      /*c_mod=*/(short)0, c, /*reuse_a=*/false, /*reuse_b=*/false);
}

// ---------------------------------------------------------------------------
// Generic batched GEMM:  Y[bat][m,n] = sum_k X[bat][m,k] * W[bat][n,k]
//                        (+ bias[n]) (ReLU?) (+ res[bat][m,n])
// K is a template constant -> all loads use immediate offsets off two base
// pointers (single s_clause per k-step, no per-iteration address VALU).
// Register-blocked 16(M) x 64(N) per wave, software-pipelined one k-step deep
// so the 4 WMMAs of step k overlap the 5 loads of step k+4.
// Grid: (N/64, M/64, batch). Block: 128 (4 waves, each one 16-row m-subtile).
// ---------------------------------------------------------------------------
template <int K>
__global__ __launch_bounds__(128) void stndt_gemm_xwt(
    const float* __restrict__ X, const float* __restrict__ W,
    const float* __restrict__ bias, const float* __restrict__ res,
    float* __restrict__ Y,
    int M, int N,
    long sX, long sW, long sY, int relu)
{
  const int lane = threadIdx.x & 31;
  const int wave = threadIdx.x >> 5;
  const int n0 = blockIdx.x * 64;
  const int m0 = (blockIdx.y * 4 + wave) * 16;
  const int bat = blockIdx.z;

  const int r  = lane & 15;          // A row (m) / B col (n) within tile
  const int kk = (lane >> 4) << 1;   // k-pair select: lanes 0-15 -> k0,k1 ; 16-31 -> k2,k3

  const float* xp = X + (long)bat * sX + (long)(m0 + r) * K + kk;
  const float* wp = W + (long)bat * sW + (long)(n0 + r) * K + kk;

  v8f acc[4] = {{}, {}, {}, {}};

  // prologue: load k-step 0
  v2f a_cur = *(const v2f*)(xp);
  v2f b_cur[4];
#pragma unroll
  for (int j = 0; j < 4; ++j) b_cur[j] = *(const v2f*)(wp + j * 16 * K);

#pragma unroll 4
  for (int k = 0; k < K - 4; k += 4) {
    // issue next k-step's loads (immediate offsets) before this step's math
    v2f a_nxt = *(const v2f*)(xp + k + 4);
    v2f b_nxt[4];
#pragma unroll
    for (int j = 0; j < 4; ++j) b_nxt[j] = *(const v2f*)(wp + j * 16 * K + k + 4);
#pragma unroll
    for (int j = 0; j < 4; ++j) acc[j] = wmma_f32_k4(a_cur, b_cur[j], acc[j]);
    a_cur = a_nxt;
#pragma unroll
    for (int j = 0; j < 4; ++j) b_cur[j] = b_nxt[j];
  }
  // epilogue: last k-step
#pragma unroll
  for (int j = 0; j < 4; ++j) acc[j] = wmma_f32_k4(a_cur, b_cur[j], acc[j]);

  float* Yb = Y + (long)bat * sY;
  const float* Rb = res ? (res + (long)bat * sY) : nullptr;
#pragma unroll
  for (int j = 0; j < 4; ++j) {
    const int col = n0 + j * 16 + (lane & 15);
    const float bv = bias ? bias[col] : 0.0f;
#pragma unroll
    for (int i = 0; i < 8; ++i) {
      const int row = m0 + i + ((lane >> 4) << 3);
      float v = acc[j][i] + bv;
      if (relu) v = fmaxf(v, 0.0f);
      if (Rb) v += Rb[(long)row * N + col];
      Yb[(long)row * N + col] = v;
    }
  }
}

// ---------------------------------------------------------------------------
// LayerNorm over last dim (length Nn), one wave per row.
// ---------------------------------------------------------------------------
__global__ __launch_bounds__(256) void stndt_ln(
    const float* __restrict__ X, const float* __restrict__ g,
    const float* __restrict__ b, float* __restrict__ Y, int Nn, int rows)
{
  const int lane = threadIdx.x & 31;
  const int wave = threadIdx.x >> 5;
  const int row = blockIdx.x * 8 + wave;
  if (row >= rows) return;
  const float* x = X + (long)row * Nn;
  float s = 0.f, ss = 0.f;
  for (int i = lane; i < Nn; i += 32) { float v = x[i]; s += v; ss += v * v; }
  for (int o = 16; o > 0; o >>= 1) { s += __shfl_xor(s, o); ss += __shfl_xor(ss, o); }
  const float m = s / (float)Nn;
  const float var = ss / (float)Nn - m * m;
  const float rst = rsqrtf(var + 1e-6f);
  float* y = Y + (long)row * Nn;
  for (int i = lane; i < Nn; i += 32) y[i] = (x[i] - m) * rst * g[i] + b[i];
}

// ---------------------------------------------------------------------------
// Temporal causal attention, per (b,h,t-tile of 16). 128 threads (4 waves).
// Q/K/V layout: [B,T, H*32] head-concat (as produced by the QKV GEMMs).
// O written in the same layout => concat-heads for the Wo GEMM.
// ---------------------------------------------------------------------------
__global__ __launch_bounds__(128) void stndt_attn(
    const float* __restrict__ Q, const float* __restrict__ Km,
    const float* __restrict__ V, float* __restrict__ O)
{
  __shared__ float sc[16 * 512];
  const int lane = threadIdx.x & 31;
  const int wave = threadIdx.x >> 5;
  const int t0 = blockIdx.x * 16;
  const int h  = blockIdx.y;
  const int b  = blockIdx.z;
  const long base = (long)b * 512 * 256 + h * 32;   // + t*256 + d

  const int r  = lane & 15;
  const int kk = (lane >> 4) << 1;
  const float* qrow = Q + base + (long)(t0 + r) * 256 + kk;

  // scores[t, s] = q . k * rsqrt(32), causal-masked
  for (int st = wave; st < 32; st += 4) {
    const int s0 = st * 16;
    const float* krow = Km + base + (long)(s0 + r) * 256 + kk;
    v8f acc = {};
#pragma unroll
    for (int k = 0; k < 32; k += 4) {
      v2f a  = *(const v2f*)(qrow + k);
      v2f bb = *(const v2f*)(krow + k);
      acc = wmma_f32_k4(a, bb, acc);
    }
    const int col = s0 + (lane & 15);
#pragma unroll
    for (int i = 0; i < 8; ++i) {
      const int rr = i + ((lane >> 4) << 3);
      float v = acc[i] * 0.17677669529663687f;      // 1/sqrt(32)
      if (col > t0 + rr) v = -__builtin_inff();     // causal mask
      sc[rr * 512 + col] = v;
    }
  }
  __syncthreads();

  // row-wise softmax over 512 (wave per row)
  for (int rr = wave; rr < 16; rr += 4) {
    float* srow = sc + rr * 512;
    float mx = -__builtin_inff();
    for (int i = lane; i < 512; i += 32) mx = fmaxf(mx, srow[i]);
    for (int o = 16; o > 0; o >>= 1) mx = fmaxf(mx, __shfl_xor(mx, o));
    float sum = 0.f;
    for (int i = lane; i < 512; i += 32) { float e = __expf(srow[i] - mx); srow[i] = e; sum += e; }
    for (int o = 16; o > 0; o >>= 1) sum += __shfl_xor(sum, o);
    const float inv = 1.0f / sum;
    for (int i = lane; i < 512; i += 32) srow[i] *= inv;
  }
  __syncthreads();

  // O(16x32) = P(16x512) @ V(512x32); waves 0,1 handle the two 16-wide halves
  if (wave < 2) {
    const int n0 = wave * 16;
    const float* arow = sc + (lane & 15) * 512 + kk;
    const float* vcol = V + base + (long)kk * 256 + n0 + (lane & 15);
    v8f acc = {};
#pragma unroll 4
    for (int k = 0; k < 512; k += 4) {
      v2f a; a.x = arow[k]; a.y = arow[k + 1];
      v2f bb; bb.x = vcol[(long)k * 256]; bb.y = vcol[(long)k * 256 + 256];
      acc = wmma_f32_k4(a, bb, acc);
    }
    const int d = n0 + (lane & 15);
#pragma unroll
    for (int i = 0; i < 8; ++i) {
      const int rr = i + ((lane >> 4) << 3);
      O[base + (long)(t0 + rr) * 256 + d] = acc[i];
    }
  }
}

// ---------------------------------------------------------------------------
// Spatial attention weights, per (b, c-tile of 16): mean_h softmax(QsKs^T/8).
// Qs/Ks layout: [B, C, H*64]. SW written as [B, C, C] (second model output).
// ---------------------------------------------------------------------------
__global__ __launch_bounds__(128) void stndt_spatial(
    const float* __restrict__ Qs, const float* __restrict__ Ks,
    float* __restrict__ SW)
{
  __shared__ float sc[16 * 256];
  __shared__ float accsm[16 * 256];
  const int lane = threadIdx.x & 31;
  const int wave = threadIdx.x >> 5;
  const int c0 = blockIdx.x * 16;
  const int b  = blockIdx.y;
  const int r  = lane & 15;
  const int kk = (lane >> 4) << 1;

  for (int i = threadIdx.x; i < 16 * 256; i += 128) accsm[i] = 0.f;
  __syncthreads();

  for (int h = 0; h < 8; ++h) {
    const long base = (long)b * 256 * 512 + h * 64;
    const float* qrow = Qs + base + (long)(c0 + r) * 512 + kk;
    for (int et = wave; et < 16; et += 4) {
      const int e0 = et * 16;
      const float* krow = Ks + base + (long)(e0 + r) * 512 + kk;
      v8f acc = {};
#pragma unroll
      for (int k = 0; k < 64; k += 4) {
        v2f a  = *(const v2f*)(qrow + k);
        v2f bb = *(const v2f*)(krow + k);
        acc = wmma_f32_k4(a, bb, acc);
      }
      const int col = e0 + (lane & 15);
#pragma unroll
      for (int i = 0; i < 8; ++i) {
        const int rr = i + ((lane >> 4) << 3);
        sc[rr * 256 + col] = acc[i] * 0.125f;       // 1/sqrt(64)
      }
    }
    __syncthreads();
    for (int rr = wave; rr < 16; rr += 4) {
      float* srow = sc + rr * 256;
      float mx = -__builtin_inff();
      for (int i = lane; i < 256; i += 32) mx = fmaxf(mx, srow[i]);
      for (int o = 16; o > 0; o >>= 1) mx = fmaxf(mx, __shfl_xor(mx, o));
      float sum = 0.f;
      for (int i = lane; i < 256; i += 32) { float e = __expf(srow[i] - mx); srow[i] = e; sum += e; }
      for (int o = 16; o > 0; o >>= 1) sum += __shfl_xor(sum, o);
      const float inv = 1.0f / sum;
      for (int i = lane; i < 256; i += 32) srow[i] *= inv;
    }
    __syncthreads();
    for (int i = threadIdx.x; i < 16 * 256; i += 128) accsm[i] += sc[i];
    __syncthreads();
  }

  for (int i = threadIdx.x; i < 16 * 256; i += 128) {
    const int rr = i >> 8, e = i & 255;
    SW[((long)b * 256 + c0 + rr) * 256 + e] = accsm[i] * 0.125f;  // mean over 8 heads
  }
}

// ---------------------------------------------------------------------------
extern "C" void kernel_launch(void* const* d_in, const int* in_sizes, int n_in,
                              void* d_out, int out_size, void* d_ws, size_t ws_size,
                              hipStream_t stream) {
  (void)in_sizes; (void)n_in; (void)out_size; (void)ws_size;
  const float* x_T    = (const float*)d_in[0];
  const float* x_S    = (const float*)d_in[1];
  const float* Wq_t   = (const float*)d_in[2];
  const float* Wk_t   = (const float*)d_in[3];
  const float* Wv_t   = (const float*)d_in[4];
  const float* Wo     = (const float*)d_in[5];
  const float* Wq_s   = (const float*)d_in[6];
  const float* Wk_s   = (const float*)d_in[7];
  const float* ff1_w1 = (const float*)d_in[8];
  const float* ff1_b1 = (const float*)d_in[9];
  const float* ff1_w2 = (const float*)d_in[10];
  const float* ff1_b2 = (const float*)d_in[11];
  const float* ff2_w1 = (const float*)d_in[12];
  const float* ff2_b1 = (const float*)d_in[13];
  const float* ff2_w2 = (const float*)d_in[14];
  const float* ff2_b2 = (const float*)d_in[15];
  const float* t_ln1_w = (const float*)d_in[16];
  const float* t_ln1_b = (const float*)d_in[17];
  const float* t_ln2_w = (const float*)d_in[18];
  const float* t_ln2_b = (const float*)d_in[19];
  const float* s_ln1_w = (const float*)d_in[20];
  const float* s_ln1_b = (const float*)d_in[21];
  const float* fus_ln_w = (const float*)d_in[22];
  const float* fus_ln_b = (const float*)d_in[23];

  // B=16, T=512, C=256, H=8; BT = 8192 rows.
  float* A  = (float*)d_ws;       // 2M floats each for A..D, 8M for E (64 MB total)
  float* Bu = A  + 2097152;
  float* Cu = Bu + 2097152;
  float* Du = Cu + 2097152;
  float* Eu = Du + 2097152;
  float* out = (float*)d_out;
  float* SW  = out + 2097152;     // second output region: [B,C,C]

  dim3 blk128(128), blk256(256);

  // 1) h = LN(x_T)                            -> A [8192,256]
  stndt_ln<<<dim3(1024), blk256, 0, stream>>>(x_T, t_ln1_w, t_ln1_b, A, 256, 8192);
  // 2) Q,K,V = h @ W{q,k,v}_t^T               -> Bu,Cu,Du [8192,256]
  stndt_gemm_xwt<256><<<dim3(4, 128, 1), blk128, 0, stream>>>(A, Wq_t, nullptr, nullptr, Bu, 8192, 256, 0, 0, 0, 0);
  stndt_gemm_xwt<256><<<dim3(4, 128, 1), blk128, 0, stream>>>(A, Wk_t, nullptr, nullptr, Cu, 8192, 256, 0, 0, 0, 0);
  stndt_gemm_xwt<256><<<dim3(4, 128, 1), blk128, 0, stream>>>(A, Wv_t, nullptr, nullptr, Du, 8192, 256, 0, 0, 0, 0);
  // 3) causal MHA                             -> A (head-concat [8192,256])
  stndt_attn<<<dim3(32, 8, 16), blk128, 0, stream>>>(Bu, Cu, Du, A);
  // 4) x = x_T + o @ Wo^T                     -> Bu
  stndt_gemm_xwt<256><<<dim3(4, 128, 1), blk128, 0, stream>>>(A, Wo, nullptr, x_T, Bu, 8192, 256, 0, 0, 0, 0);
  // 5) h2 = LN(x)                             -> A
  stndt_ln<<<dim3(1024), blk256, 0, stream>>>(Bu, t_ln2_w, t_ln2_b, A, 256, 8192);
  // 6) mid = relu(h2 @ ff1_w1^T + b1)         -> Eu [8192,1024]
  stndt_gemm_xwt<256><<<dim3(16, 128, 1), blk128, 0, stream>>>(A, ff1_w1, ff1_b1, nullptr, Eu, 8192, 1024, 0, 0, 0, 1);
  // 7) temporal_out = x + mid @ ff1_w2^T + b2 -> Cu
  stndt_gemm_xwt<1024><<<dim3(4, 128, 1), blk128, 0, stream>>>(Eu, ff1_w2, ff1_b2, Bu, Cu, 8192, 256, 0, 0, 0, 0);
  // 8) hs = LN(x_S over T)                    -> A [4096,512]
  stndt_ln<<<dim3(512), blk256, 0, stream>>>(x_S, s_ln1_w, s_ln1_b, A, 512, 4096);
  // 9) Qs,Ks = hs @ W{q,k}_s^T                -> Bu,Du [4096,512]
  stndt_gemm_xwt<512><<<dim3(8, 64, 1), blk128, 0, stream>>>(A, Wq_s, nullptr, nullptr, Bu, 4096, 512, 0, 0, 0, 0);
  stndt_gemm_xwt<512><<<dim3(8, 64, 1), blk128, 0, stream>>>(A, Wk_s, nullptr, nullptr, Du, 4096, 512, 0, 0, 0, 0);
  // 10) SW = mean_h softmax(QsKs^T/8)         -> d_out second region [16,256,256]
  stndt_spatial<<<dim3(16, 16, 1), blk128, 0, stream>>>(Bu, Du, SW);
  // 11) x2 = TO + einsum('bce,bte->btc')      -> A (batched GEMM, W = SW per batch)
  stndt_gemm_xwt<256><<<dim3(4, 8, 16), blk128, 0, stream>>>(Cu, SW, nullptr, Cu, A, 512, 256, 131072, 65536, 131072, 0);
  // 12) h3 = LN(x2)                           -> Bu
  stndt_ln<<<dim3(1024), blk256, 0, stream>>>(A, fus_ln_w, fus_ln_b, Bu, 256, 8192);
  // 13) mid2 = relu(h3 @ ff2_w1^T + b1)       -> Eu
  stndt_gemm_xwt<256><<<dim3(16, 128, 1), blk128, 0, stream>>>(Bu, ff2_w1, ff2_b1, nullptr, Eu, 8192, 1024, 0, 0, 0, 1);
  // 14) out = x2 + mid2 @ ff2_w2^T + b2       -> d_out first region
  stndt_gemm_xwt<1024><<<dim3(4, 128, 1), blk128, 0, stream>>>(Eu, ff2_w2, ff2_b2, A, out, 8192, 256, 0, 0, 0, 0);
}